// BIMM2D_6416681140899
// MI455X (gfx1250) — compile-verified
//
#include <hip/hip_runtime.h>
#include <math.h>

#define L2E 1.44269504088896f
#define LN2 0.69314718055995f
#define LOGN 4.15888308335967f   // ln(64)

// ---- workspace float layout ----
#define PKN_OFF  0      // 384 x float4 {In, cn*L2E, (2*Gk/s2)*L2E, 0} = 1536 floats
#define LW_OFF   1536   // log_softmax(W), 10 floats
#define SC_OFF   1546   // 5 scalar constants
#define I_OFF    1551   // I[0..3]
#define PART_OFF 1568   // per-block partial sums

#define NT 256

__device__ __forceinline__ float fast_log(float x) {
  return __builtin_amdgcn_logf(x) * LN2;   // v_log_f32 is log2
}

// ---------------- kernel 1: per-(k,n) constants + log_softmax(W) ----------------
__global__ void bimm_precompute(const float* __restrict__ eps,
                                const float* __restrict__ I,
                                const float* __restrict__ W,
                                const float* __restrict__ sigma_b,
                                const float* __restrict__ sigma_n,
                                const float* __restrict__ dvec,
                                const float* __restrict__ rvec,
                                float* __restrict__ ws) {
  const int i = threadIdx.x;
  const float sb  = sigma_b[0];
  const float sn  = sigma_n[0];
  const float dd  = dvec[0];
  const float rho = tanhf(rvec[0]);
  const float sr  = sn * sqrtf(1.0f - rho);
  const float s2  = sn * sn * (1.0f - rho);
  const float inv_s2 = 1.0f / s2;

  if (i < 384) {
    const int k = i >> 6;
    const int IAk[6] = {0, 0, 0, 1, 1, 2};
    const int IBk[6] = {1, 2, 3, 2, 3, 3};
    const float Ia = I[IAk[k]];
    const float Ib = I[IBk[k]];
    const float x  = eps[i] * (2.0f * dd * sb) - dd * sb;
    const float In = (erff(x / (1.41421356237f * sb)) + 1.0f) * 0.5f * (Ib - Ia) + Ia;
    const float q  = 2.0f * (In - Ia) / (Ib - Ia) - 1.0f;
    const float ei = erfinvf(q);
    const float Gk = (Ib - Ia) / sqrtf(6.28318530718f * sb * sb) * expf(-ei * ei);
    const float cn = -logf(Gk) - Gk * Gk * inv_s2;     // -(log Gk + Gk^2/s2)
    float4 p;
    p.x = In;
    p.y = cn * L2E;                    // log2-domain additive constant
    p.z = 2.0f * Gk * inv_s2 * L2E;    // zl = v * p.z  (== z*log2e)
    p.w = 0.0f;
    ((float4*)ws)[i] = p;
  } else if (i == 384) {
    // log_softmax over W (10 entries)
    float mx = W[0];
    for (int j = 1; j < 10; ++j) mx = fmaxf(mx, W[j]);
    float s = 0.0f;
    for (int j = 0; j < 10; ++j) s += expf(W[j] - mx);
    const float lse = mx + logf(s);
    for (int j = 0; j < 10; ++j) ws[LW_OFF + j] = W[j] - lse;
    // base_const = -ln sn - 0.5 ln(2pi) - ln sr - 0.5 ln(pi)
    ws[SC_OFF + 0] = -logf(sn) - 0.91893853320467f - logf(sr) - 0.57236494292470f;
    // const_int = ln2 - ln(Gamma(1.5)) - 3 ln sr - ln sn - 0.5 ln(2pi)
    ws[SC_OFF + 1] = 0.69314718056f + 0.12078223764f - 3.0f * logf(sr)
                     - logf(sn) - 0.91893853320467f;
    ws[SC_OFF + 2] = inv_s2;
    ws[SC_OFF + 3] = -0.5f / (sn * sn);          // natural-log domain (interior)
    ws[SC_OFF + 4] = (-0.5f / (sn * sn)) * L2E;  // log2 domain (interface lu)
    for (int j = 0; j < 4; ++j) ws[I_OFF + j] = I[j];
  }
}

// ---------------- kernel 2: per-point NLL, per-block partial sums ----------------
typedef float v2f __attribute__((ext_vector_type(2)));
typedef float v8f __attribute__((ext_vector_type(8)));

__global__ void bimm_main(const float* __restrict__ u,
                          const float* __restrict__ v,
                          const float* __restrict__ ws,
                          float* __restrict__ partials,
                          int M) {
  __shared__ float4 sP[384];
  __shared__ float swave[NT / 32];
  const int tid = threadIdx.x;

  const float4* wsP = (const float4*)ws;
  for (int i = tid; i < 384; i += NT) sP[i] = wsP[i];

  float lw[10];
#pragma unroll
  for (int j = 0; j < 10; ++j) lw[j] = ws[LW_OFF + j];
  const float base_const = ws[SC_OFF + 0];
  const float const_int  = ws[SC_OFF + 1];
  const float inv_s2     = ws[SC_OFF + 2];
  const float nhis       = ws[SC_OFF + 3];
  const float c_du       = ws[SC_OFF + 4];
  float Iv[4];
#pragma unroll
  for (int j = 0; j < 4; ++j) Iv[j] = ws[I_OFF + j];
  __syncthreads();

  float acc = 0.0f;
  const int stride = gridDim.x * NT;
  for (int m = blockIdx.x * NT + tid; m < M; m += stride) {
    const float um  = u[m];
    const float vm  = v[m];
    const float lvn = fast_log(vm);
    const float v2s = vm * vm * inv_s2;
    const float pm   = base_const + lvn - v2s;          // natural-log per-point const
    const float pint = const_int + 2.0f * lvn - v2s;

    float rows[10];
#pragma unroll
    for (int p = 0; p < 4; ++p) {                        // interior phases
      const float du = um - Iv[p];
      rows[p] = fmaf(du * du, nhis, pint);
    }

    for (int k = 0; k < 6; ++k) {                        // interface pairs
      const float4* pk = &sP[k * 64];
      float S = 0.0f;
#pragma unroll 4
      for (int n = 0; n < 64; ++n) {
        const float4 p = pk[n];                          // ds_load_b128 broadcast
        const float du = um - p.x;
        const float zl = vm * p.z;                       // z * log2e
        const float a  = fmaf(du * du, c_du, p.y) + zl;  // (t - pm - log1p)*log2e
        const float e  = __builtin_amdgcn_exp2f(a);      // v_exp_f32
        const float w  = __builtin_amdgcn_exp2f(-(zl + zl)); // exp(-2z)
        S = fmaf(e, 1.0f - w, S);                        // exp(t - pm) summed
      }
      rows[4 + k] = pm - LOGN + fast_log(S);
    }

    // logsumexp over 10 mixture components with log-weights
    float vals[10];
    float mx = -3.4e38f;
#pragma unroll
    for (int j = 0; j < 10; ++j) {
      vals[j] = rows[j] + lw[j];
      mx = fmaxf(mx, vals[j]);
    }
    float ss = 0.0f;
#pragma unroll
    for (int j = 0; j < 10; ++j) ss += __builtin_amdgcn_exp2f((vals[j] - mx) * L2E);
    acc += mx + fast_log(ss);
  }

  // ---- block reduction: one V_WMMA_F32_16X16X4_F32 reduces each wave's lanes ----
  // A(16x4) holds the 32 lane partials (K0 <- lanes 0..15, K2 <- lanes 16..31),
  // B = all-ones, so D[i][j] = acc_i + acc_{i+16} exactly in f32. Summing a
  // lane's 8 D VGPRs covers one full D column after the cross-half exchange,
  // yielding sum(acc) in every lane.
  v2f A; A[0] = acc;  A[1] = 0.0f;
  v2f B; B[0] = 1.0f; B[1] = 1.0f;
  v8f C = {};
  C = __builtin_amdgcn_wmma_f32_16x16x4_f32(false, A, false, B, (short)0, C, false, false);
  float hs = ((C[0] + C[1]) + (C[2] + C[3])) + ((C[4] + C[5]) + (C[6] + C[7]));
  hs += __shfl_xor(hs, 16, 32);

  const int lane = tid & 31;
  const int wave = tid >> 5;
  if (lane == 0) swave[wave] = hs;
  __syncthreads();
  if (tid == 0) {
    float bsum = 0.0f;
    for (int w2 = 0; w2 < NT / 32; ++w2) bsum += swave[w2];
    partials[blockIdx.x] = bsum;
  }
}

// ---------------- kernel 3: deterministic cross-block reduction ----------------
__global__ void bimm_finalize(const float* __restrict__ partials, int nblocks,
                              float* __restrict__ out, float invM) {
  __shared__ float sm[NT];
  const int tid = threadIdx.x;
  float s = 0.0f;
  for (int i = tid; i < nblocks; i += NT) s += partials[i];
  sm[tid] = s;
  __syncthreads();
  for (int off = NT / 2; off > 0; off >>= 1) {
    if (tid < off) sm[tid] += sm[tid + off];
    __syncthreads();
  }
  if (tid == 0) out[0] = -sm[0] * invM;
}

extern "C" void kernel_launch(void* const* d_in, const int* in_sizes, int n_in,
                              void* d_out, int out_size, void* d_ws, size_t ws_size,
                              hipStream_t stream) {
  (void)n_in; (void)out_size; (void)ws_size;
  const float* u   = (const float*)d_in[0];
  const float* v   = (const float*)d_in[1];
  const float* eps = (const float*)d_in[2];
  const float* I   = (const float*)d_in[3];
  const float* W   = (const float*)d_in[4];
  const float* sb  = (const float*)d_in[5];
  const float* sn  = (const float*)d_in[6];
  const float* dd  = (const float*)d_in[7];
  const float* rr  = (const float*)d_in[8];
  const int M = in_sizes[0];

  float* ws = (float*)d_ws;
  float* partials = ws + PART_OFF;
  const int nblocks = (M + NT - 1) / NT;

  bimm_precompute<<<1, 448, 0, stream>>>(eps, I, W, sb, sn, dd, rr, ws);
  bimm_main<<<nblocks, NT, 0, stream>>>(u, v, ws, partials, M);
  bimm_finalize<<<1, NT, 0, stream>>>(partials, nblocks, (float*)d_out,
                                      1.0f / (float)M);
}